// SimpleMambaLMHeadModel_13030930776216
// MI455X (gfx1250) — compile-verified
//
#include <hip/hip_runtime.h>
#include <hip/hip_bf16.h>
#include <math.h>

typedef __attribute__((ext_vector_type(2))) float v2f;
typedef __attribute__((ext_vector_type(8))) float v8f;
typedef __attribute__((ext_vector_type(4))) int   v4i;

#define D_MODEL 768
#define N_LAYER 2
#define VOCAB   50257
#define D_STATE 16
#define D_CONV  4
#define D_INNER 1536
#define DT_RANK 48
#define BB      2
#define LL      1024
#define NTOK    (BB*LL)      /* 2048 */

/* -------- async global->LDS support (guarded; falls back to sync copy) ---- */
#if defined(__has_builtin)
#  if __has_builtin(__builtin_amdgcn_global_load_async_to_lds_b128)
#    define HAVE_ASYNC_B128 1
#  endif
#  if __has_builtin(__builtin_amdgcn_s_wait_asynccnt)
#    define HAVE_WAIT_ASYNC 1
#  endif
#endif

__device__ __forceinline__ void async_copy16(const float* g, float* l)
{
#ifdef HAVE_ASYNC_B128
    /* builtin expects v4i* params (probe-confirmed by round-2 diagnostic);
       order follows the global.load.lds convention: (global src, lds dst). */
    __builtin_amdgcn_global_load_async_to_lds_b128(
        (v4i*)(void*)g, (v4i*)l, 0, 0);
#else
    *reinterpret_cast<float4*>(l) = *reinterpret_cast<const float4*>(g);
#endif
}

__device__ __forceinline__ void wait_async()
{
#ifdef HAVE_ASYNC_B128
#  ifdef HAVE_WAIT_ASYNC
    __builtin_amdgcn_s_wait_asynccnt(0);
#  else
    asm volatile("s_wait_asynccnt 0x0" ::: "memory");
#  endif
#endif
}

/* ---------------- WMMA f32 GEMM: C[M,N] = A[M,K] * W[N,K]^T ---------------- */
/* 128 threads = 4 waves. Block tile 64x128, wave tile 32x64 (2x4 frags).
   K in chunks of 32, double-buffered LDS fed by async b128 copies when the
   tile is fully in-bounds (block-uniform decision -> EXEC all-1s at wmma).
   LDS row stride 36 floats: 144B rows keep b128 dests 16B-aligned and are
   bank-conflict-free across 16 consecutive rows (gcd(36,64)=4 -> 16-cycle). */

#define MB 64
#define NB 128
#define KB 32
#define LDSS 36

#define EPI_NONE 0
#define EPI_SOFTPLUS_BIAS 1

__global__ __launch_bounds__(128)
void gemm_wmma_f32(const float* __restrict__ A, int lda,
                   const float* __restrict__ W, int ldb,
                   float* __restrict__ C, int ldc,
                   const float* __restrict__ bias,
                   int M, int N, int K, int epi)
{
    __shared__ float As[2][MB][LDSS];
    __shared__ float Bs[2][NB][LDSS];

    const int tid   = threadIdx.x;
    const int lane  = tid & 31;
    const int wave  = tid >> 5;
    const int lmod  = lane & 15;
    const int lhalf = lane >> 4;          /* 0: lanes 0-15, 1: lanes 16-31 */
    const int wm    = (wave & 1) * 32;    /* wave row offset inside tile */
    const int wn    = (wave >> 1) * 64;   /* wave col offset inside tile */

    const int m0 = blockIdx.y * MB;
    const int n0 = blockIdx.x * NB;

    v8f acc[2][4] = {};

    const int nchunk = (K + KB - 1) / KB;

    auto stage = [&](int c, int buf) {
        const int k0 = c * KB;
        const bool full = (m0 + MB <= M) && (n0 + NB <= N) && (k0 + KB <= K);
        if (full) {
            #pragma unroll
            for (int it = 0; it < 4; ++it) {               /* A: 64x32 */
                int idx = tid + 128 * it;
                int row = idx >> 3;
                int kq  = (idx & 7) << 2;
                async_copy16(A + (size_t)(m0 + row) * lda + k0 + kq,
                             &As[buf][row][kq]);
            }
            #pragma unroll
            for (int it = 0; it < 8; ++it) {               /* B: 128x32 */
                int idx = tid + 128 * it;
                int row = idx >> 3;
                int kq  = (idx & 7) << 2;
                async_copy16(W + (size_t)(n0 + row) * ldb + k0 + kq,
                             &Bs[buf][row][kq]);
            }
        } else {
            #pragma unroll
            for (int it = 0; it < 4; ++it) {
                int idx = tid + 128 * it;
                int row = idx >> 3;
                int kq  = (idx & 7) << 2;
                int gm  = m0 + row;
                #pragma unroll
                for (int u = 0; u < 4; ++u) {
                    int gk = k0 + kq + u;
                    float v = 0.f;
                    if (gm < M && gk < K) v = A[(size_t)gm * lda + gk];
                    As[buf][row][kq + u] = v;
                }
            }
            #pragma unroll
            for (int it = 0; it < 8; ++it) {
                int idx = tid + 128 * it;
                int row = idx >> 3;
                int kq  = (idx & 7) << 2;
                int gn  = n0 + row;
                #pragma unroll
                for (int u = 0; u < 4; ++u) {
                    int gk = k0 + kq + u;
                    float v = 0.f;
                    if (gn < N && gk < K) v = W[(size_t)gn * ldb + gk];
                    Bs[buf][row][kq + u] = v;
                }
            }
        }
    };

    stage(0, 0);

    for (int c = 0; c < nchunk; ++c) {
        const int buf = c & 1;
        wait_async();          /* my async writes into buf are done */
        __syncthreads();       /* everyone's writes visible; buf^1 free */
        if (c + 1 < nchunk) stage(c + 1, buf ^ 1);

        float (*Ab)[LDSS] = As[buf];
        float (*Bb)[LDSS] = Bs[buf];
        #pragma unroll
        for (int kk = 0; kk < KB; kk += 4) {
            int ka = kk + 2 * lhalf;  /* lanes 0-15 hold K={kk,kk+1}, 16-31 K={kk+2,kk+3} */
            v2f af[2], bf[4];
            af[0].x = Ab[wm +  0 + lmod][ka];
            af[0].y = Ab[wm +  0 + lmod][ka + 1];
            af[1].x = Ab[wm + 16 + lmod][ka];
            af[1].y = Ab[wm + 16 + lmod][ka + 1];
            #pragma unroll
            for (int j = 0; j < 4; ++j) {
                bf[j].x = Bb[wn + 16 * j + lmod][ka];
                bf[j].y = Bb[wn + 16 * j + lmod][ka + 1];
            }
            #pragma unroll
            for (int i = 0; i < 2; ++i)
                #pragma unroll
                for (int j = 0; j < 4; ++j)
                    acc[i][j] = __builtin_amdgcn_wmma_f32_16x16x4_f32(
                        false, af[i], false, bf[j], (short)0, acc[i][j],
                        false, false);
        }
        __syncthreads();       /* done reading buf before it is restaged */
    }

    /* ---- epilogue. C/D layout: n = lane&15, m = r + 8*(lane>>4) ---- */
    #pragma unroll
    for (int i = 0; i < 2; ++i) {
        #pragma unroll
        for (int j = 0; j < 4; ++j) {
            int n  = n0 + wn + 16 * j + lmod;
            int mb = m0 + wm + 16 * i + 8 * lhalf;
            #pragma unroll
            for (int r = 0; r < 8; ++r) {
                int m = mb + r;
                if (m < M && n < N) {
                    float v = acc[i][j][r];
                    if (epi == EPI_SOFTPLUS_BIAS) {
                        v += bias[n];
                        v = (v > 20.f) ? v : log1pf(__expf(v));
                    }
                    C[(size_t)m * ldc + n] = v;
                }
            }
        }
    }
}

/* ---------------- embedding gather ---------------- */
__global__ __launch_bounds__(256)
void embed_kernel(const int* __restrict__ ids, const float* __restrict__ emb,
                  float* __restrict__ x)
{
    int t = blockIdx.x;
    const float* src = emb + (size_t)ids[t] * D_MODEL;
    float* dst = x + (size_t)t * D_MODEL;
    for (int d = threadIdx.x; d < D_MODEL; d += 256) dst[d] = src[d];
}

/* ---------------- causal depthwise conv (width 4) + SiLU ---------------- */
__global__ __launch_bounds__(256)
void conv_silu_kernel(const float* __restrict__ xz,
                      const float* __restrict__ cw,
                      const float* __restrict__ cb,
                      float* __restrict__ xc)
{
    int idx = blockIdx.x * 256 + threadIdx.x;
    if (idx >= NTOK * D_INNER) return;
    int d = idx % D_INNER;
    int t = idx / D_INNER;
    int l = t % LL;
    float s = cb[d];
    #pragma unroll
    for (int k = 0; k < D_CONV; ++k) {
        int ls = l - (D_CONV - 1) + k;
        if (ls >= 0)
            s += xz[(size_t)(t - (l - ls)) * (2 * D_INNER) + d] * cw[d * D_CONV + k];
    }
    xc[idx] = s / (1.f + __expf(-s));   /* silu */
}

/* ---------------- selective scan: one thread per (b, d) channel ---------------- */
__global__ __launch_bounds__(256)
void scan_kernel(const float* __restrict__ dt,
                 const float* __restrict__ xc,
                 const float* __restrict__ dbc,
                 const float* __restrict__ A_log,
                 float* __restrict__ ys)
{
    int idx = blockIdx.x * 256 + threadIdx.x;
    if (idx >= BB * D_INNER) return;
    int d = idx % D_INNER;
    int b = idx / D_INNER;

    float Ad[D_STATE], h[D_STATE];
    #pragma unroll
    for (int n = 0; n < D_STATE; ++n) {
        Ad[n] = -__expf(A_log[d * D_STATE + n]);
        h[n] = 0.f;
    }
    for (int l = 0; l < LL; ++l) {
        int t = b * LL + l;
        float dtv = dt[(size_t)t * D_INNER + d];
        float xv  = xc[(size_t)t * D_INNER + d];
        const float* bc = dbc + (size_t)t * (DT_RANK + 2 * D_STATE) + DT_RANK;
        if (l + 8 < LL) {   /* hide serial-loop latency: prefetch 8 steps ahead */
            __builtin_prefetch(&dt[(size_t)(t + 8) * D_INNER + d], 0, 1);
            __builtin_prefetch(&xc[(size_t)(t + 8) * D_INNER + d], 0, 1);
        }
        float y = 0.f;
        #pragma unroll
        for (int n = 0; n < D_STATE; ++n) {
            float dA = __expf(dtv * Ad[n]);
            h[n] = dA * h[n] + dtv * bc[n] * xv;
            y += h[n] * bc[D_STATE + n];
        }
        ys[(size_t)t * D_INNER + d] = y;
    }
}

/* ---------------- y = (ys + Dp*xc) * silu(z) ---------------- */
__global__ __launch_bounds__(256)
void gate_kernel(const float* __restrict__ ys, const float* __restrict__ xc,
                 const float* __restrict__ xz, const float* __restrict__ Dp,
                 float* __restrict__ y)
{
    int idx = blockIdx.x * 256 + threadIdx.x;
    if (idx >= NTOK * D_INNER) return;
    int d = idx % D_INNER;
    int t = idx / D_INNER;
    float z = xz[(size_t)t * (2 * D_INNER) + D_INNER + d];
    float v = ys[idx] + Dp[d] * xc[idx];
    y[idx] = v * (z / (1.f + __expf(-z)));
}

/* ---------------- LayerNorm over D_MODEL ---------------- */
__global__ __launch_bounds__(256)
void ln_kernel(const float* __restrict__ x, const float* __restrict__ w,
               const float* __restrict__ b, float* __restrict__ o)
{
    __shared__ float red0[8], red1[8];
    int t = blockIdx.x;
    const float* xr = x + (size_t)t * D_MODEL;
    float s = 0.f, s2 = 0.f;
    for (int d = threadIdx.x; d < D_MODEL; d += 256) {
        float v = xr[d]; s += v; s2 += v * v;
    }
    for (int off = 16; off > 0; off >>= 1) {
        s  += __shfl_down(s,  off, 32);
        s2 += __shfl_down(s2, off, 32);
    }
    int lane = threadIdx.x & 31, wv = threadIdx.x >> 5;
    if (lane == 0) { red0[wv] = s; red1[wv] = s2; }
    __syncthreads();
    if (wv == 0) {
        s  = (lane < 8) ? red0[lane] : 0.f;
        s2 = (lane < 8) ? red1[lane] : 0.f;
        for (int off = 4; off > 0; off >>= 1) {
            s  += __shfl_down(s,  off, 32);
            s2 += __shfl_down(s2, off, 32);
        }
        if (lane == 0) { red0[0] = s; red1[0] = s2; }
    }
    __syncthreads();
    float mu  = red0[0] / (float)D_MODEL;
    float var = red1[0] / (float)D_MODEL - mu * mu;
    float rs  = rsqrtf(var + 1e-5f);
    for (int d = threadIdx.x; d < D_MODEL; d += 256)
        o[(size_t)t * D_MODEL + d] = (xr[d] - mu) * rs * w[d] + b[d];
}

/* ---------------- driver ---------------- */
extern "C" void kernel_launch(void* const* d_in, const int* in_sizes, int n_in,
                              void* d_out, int out_size, void* d_ws, size_t ws_size,
                              hipStream_t stream)
{
    const int*   ids   = (const int*)  d_in[0];
    const float* emb   = (const float*)d_in[1];
    const float* Wi_a  = (const float*)d_in[2];
    const float* cw_a  = (const float*)d_in[3];
    const float* cb_a  = (const float*)d_in[4];
    const float* Wx_a  = (const float*)d_in[5];
    const float* Wdt_a = (const float*)d_in[6];
    const float* bdt_a = (const float*)d_in[7];
    const float* Al_a  = (const float*)d_in[8];
    const float* Dp_a  = (const float*)d_in[9];
    const float* Wo_a  = (const float*)d_in[10];
    const float* nw    = (const float*)d_in[11];
    const float* nb    = (const float*)d_in[12];
    const float* Wh    = (const float*)d_in[13];
    float* out = (float*)d_out;

    float* ws = (float*)d_ws;
    size_t off = 0;
    float* x    = ws + off; off += (size_t)NTOK * D_MODEL;        /* 2048x768  */
    float* xz   = ws + off; off += (size_t)NTOK * 2 * D_INNER;    /* 2048x3072 */
    float* xc   = ws + off; off += (size_t)NTOK * D_INNER;        /* 2048x1536 */
    float* dbc  = ws + off; off += (size_t)NTOK * (DT_RANK + 2 * D_STATE);
    float* dt   = ws + off; off += (size_t)NTOK * D_INNER;
    float* ys   = ws + off; off += (size_t)NTOK * D_INNER;
    float* y    = ws + off; off += (size_t)NTOK * D_INNER;
    float* xn   = ws + off; off += (size_t)NTOK * D_MODEL;

    dim3 blk128(128), blk256(256);
    int elemsDI = NTOK * D_INNER;
    int gElemDI = (elemsDI + 255) / 256;

    embed_kernel<<<dim3(NTOK), blk256, 0, stream>>>(ids, emb, x);

    for (int layer = 0; layer < N_LAYER; ++layer) {
        const float* Wi  = Wi_a  + (size_t)layer * 2 * D_INNER * D_MODEL;
        const float* cw  = cw_a  + (size_t)layer * D_INNER * D_CONV;
        const float* cb  = cb_a  + (size_t)layer * D_INNER;
        const float* Wx  = Wx_a  + (size_t)layer * (DT_RANK + 2 * D_STATE) * D_INNER;
        const float* Wdt = Wdt_a + (size_t)layer * D_INNER * DT_RANK;
        const float* bdt = bdt_a + (size_t)layer * D_INNER;
        const float* Al  = Al_a  + (size_t)layer * D_INNER * D_STATE;
        const float* Dpl = Dp_a  + (size_t)layer * D_INNER;
        const float* Wo  = Wo_a  + (size_t)layer * D_MODEL * D_INNER;

        /* xz = x @ Wi^T : 2048 x 3072 x 768 */
        gemm_wmma_f32<<<dim3((2 * D_INNER + NB - 1) / NB, (NTOK + MB - 1) / MB),
                        blk128, 0, stream>>>(
            x, D_MODEL, Wi, D_MODEL, xz, 2 * D_INNER, nullptr,
            NTOK, 2 * D_INNER, D_MODEL, EPI_NONE);

        conv_silu_kernel<<<dim3(gElemDI), blk256, 0, stream>>>(xz, cw, cb, xc);

        /* dbc = xc @ Wx^T : 2048 x 80 x 1536 */
        gemm_wmma_f32<<<dim3((DT_RANK + 2 * D_STATE + NB - 1) / NB,
                             (NTOK + MB - 1) / MB), blk128, 0, stream>>>(
            xc, D_INNER, Wx, D_INNER, dbc, DT_RANK + 2 * D_STATE, nullptr,
            NTOK, DT_RANK + 2 * D_STATE, D_INNER, EPI_NONE);

        /* dt = softplus(dbc[:, :48] @ Wdt^T + bdt) : 2048 x 1536 x 48 */
        gemm_wmma_f32<<<dim3((D_INNER + NB - 1) / NB, (NTOK + MB - 1) / MB),
                        blk128, 0, stream>>>(
            dbc, DT_RANK + 2 * D_STATE, Wdt, DT_RANK, dt, D_INNER, bdt,
            NTOK, D_INNER, DT_RANK, EPI_SOFTPLUS_BIAS);

        scan_kernel<<<dim3((BB * D_INNER + 255) / 256), blk256, 0, stream>>>(
            dt, xc, dbc, Al, ys);

        gate_kernel<<<dim3(gElemDI), blk256, 0, stream>>>(ys, xc, xz, Dpl, y);

        /* x = y @ Wo^T : 2048 x 768 x 1536 */
        gemm_wmma_f32<<<dim3((D_MODEL + NB - 1) / NB, (NTOK + MB - 1) / MB),
                        blk128, 0, stream>>>(
            y, D_INNER, Wo, D_INNER, x, D_MODEL, nullptr,
            NTOK, D_MODEL, D_INNER, EPI_NONE);
    }

    ln_kernel<<<dim3(NTOK), blk256, 0, stream>>>(x, nw, nb, xn);

    /* logits = xn @ head_w^T : 2048 x 50257 x 768 */
    gemm_wmma_f32<<<dim3((VOCAB + NB - 1) / NB, (NTOK + MB - 1) / MB),
                    blk128, 0, stream>>>(
        xn, D_MODEL, Wh, D_MODEL, out, VOCAB, nullptr,
        NTOK, VOCAB, D_MODEL, EPI_NONE);

    (void)in_sizes; (void)n_in; (void)out_size; (void)ws_size;
}